// ExtractPatchLayer3_36696200577416
// MI455X (gfx1250) — compile-verified
//
#include <hip/hip_runtime.h>

// Patch extraction (im2col), k=5, pad=2.
// Input : (8,128,128,32) f32  -> 16.8 MB (fits in L2)
// Output: (8,128,128,800) f32 -> 419 MB  (store-bound; HBM floor ~19us @23.3TB/s)
// Strategy: 1 workgroup per (b,h) row; stage 5 padded input rows in LDS via
// CDNA5 async global->LDS DMA (global_load_async_to_lds_b128 + s_wait_asynccnt);
// emit a perfectly linear stream of b128 non-temporal stores.

typedef float v4f __attribute__((ext_vector_type(4)));
typedef int   v4i_gcc __attribute__((vector_size(16)));   // matches builtin param

constexpr int Bn = 8, Hn = 128, Wn = 128, Cn = 32, Kk = 5, Pp = 2;
constexpr int WPAD       = Wn + 2 * Pp;          // 132 padded columns
constexpr int LDS_ROW_F  = WPAD * Cn;            // 4224 floats per padded row
constexpr int LDS_ROW_V4 = LDS_ROW_F / 4;        // 1056 float4 per padded row
constexpr int ROW_OUT_V4 = Wn * (Kk * Kk * Cn / 4); // 25600 float4 per output row

__device__ __forceinline__ void async_load_b128(const float* gsrc, float* ldst) {
#if __has_builtin(__builtin_amdgcn_global_load_async_to_lds_b128)
  __builtin_amdgcn_global_load_async_to_lds_b128(
      (__attribute__((address_space(1))) v4i_gcc*)gsrc,
      (__attribute__((address_space(3))) v4i_gcc*)ldst,
      0, 0);
#else
  unsigned lds_off = (unsigned)(unsigned long long)ldst;
  asm volatile("global_load_async_to_lds_b128 %0, %1, off"
               :: "v"(lds_off), "v"(gsrc)
               : "memory");
#endif
}

__device__ __forceinline__ void wait_async0() {
#if __has_builtin(__builtin_amdgcn_s_wait_asynccnt)
  __builtin_amdgcn_s_wait_asynccnt(0);
#else
  asm volatile("s_wait_asynccnt 0x0" ::: "memory");
#endif
}

__global__ __launch_bounds__(256) void extract_patches_k5(
    const float* __restrict__ in, float* __restrict__ out) {
  __shared__ __align__(16) float smem[Kk * LDS_ROW_F];  // 5*132*32 f32 = 84,480 B
  v4f* s4 = (v4f*)smem;

  const int t  = threadIdx.x;              // 0..255
  const int bh = blockIdx.x;               // b*128 + h, 0..1023
  const int h  = bh & (Hn - 1);

  const v4f zero = {0.f, 0.f, 0.f, 0.f};

  // --- 1) zero the left/right column pads (cols 0,1,130,131) of all 5 rows ---
  // 5 rows * (2+2 cols) * 32 ch = 640 f32 = 160 float4
  if (t < 160) {
    int row = t >> 5, within = t & 31, half = within >> 4, j = within & 15;
    s4[row * LDS_ROW_V4 + half * 1040 + j] = zero;   // 1040 = col130*32/4
  }

  // --- 2) per tap-row: zero out-of-range rows, async-DMA valid rows to LDS ---
#pragma unroll
  for (int dy = 0; dy < Kk; ++dy) {
    int r = h - Pp + dy;
    if (r < 0 || r >= Hn) {
      // zero interior (cols 2..129): 128*32/4 = 1024 float4
      int base4 = dy * LDS_ROW_V4 + 16;
#pragma unroll
      for (int j = 0; j < 4; ++j) s4[base4 + t + j * 256] = zero;
    } else {
      // (b*H + r)*W*C = (bh - 2 + dy) * 4096 floats, contiguous 16 KB row
      const float* src = in + ((size_t)(bh - Pp + dy) << 12);
      float*       dst = smem + dy * LDS_ROW_F + 2 * Cn;  // col offset 2
#pragma unroll
      for (int j = 0; j < 4; ++j) {
        int chunk = t + j * 256;                    // 1024 x 16B chunks/row
        async_load_b128(src + chunk * 4, dst + chunk * 4);
      }
    }
  }

  wait_async0();        // this wave's async DMAs have landed in LDS
  __syncthreads();      // all waves' contributions visible (also waits DScnt)

  // --- 3) emit: out float4 index g = w*200 + tap*8 + c4  ->  store addr is
  //        rowBase + 16*g (perfectly linear b128 stream). Only the LDS gather
  //        needs the (w, dy, dx, c4) decomposition.
  v4f* out4 = (v4f*)out + (size_t)bh * ROW_OUT_V4;
#pragma unroll 5
  for (int i = 0; i < 100; ++i) {
    unsigned g   = (unsigned)(i * 256 + t);   // 0..25599
    unsigned w   = g / 200u;                  // const-div -> umulhi
    unsigned rem = g - w * 200u;
    unsigned tap = rem >> 3;
    unsigned c4  = rem & 7u;
    unsigned dy  = tap / 5u;
    unsigned dx  = tap - dy * 5u;
    v4f v = s4[(dy * WPAD + w + dx) * (Cn / 4) + c4];
    __builtin_nontemporal_store(v, out4 + g); // b128 store, TH=NT
  }
}

extern "C" void kernel_launch(void* const* d_in, const int* in_sizes, int n_in,
                              void* d_out, int out_size, void* d_ws, size_t ws_size,
                              hipStream_t stream) {
  (void)in_sizes; (void)n_in; (void)out_size; (void)d_ws; (void)ws_size;
  const float* in = (const float*)d_in[0];
  float* out = (float*)d_out;
  hipLaunchKernelGGL(extract_patches_k5, dim3(Bn * Hn), dim3(256), 0, stream,
                     in, out);
}